// Codebook_topk_81080392614187
// MI455X (gfx1250) — compile-verified
//
#include <hip/hip_runtime.h>
#include <math.h>

// ---------------- problem constants ----------------
constexpr int   DIM      = 64;
constexpr int   SIZE     = 1024;          // codebook entries
constexpr int   TOPK     = 3;
constexpr int   NVEC     = 16 * 64 * 64;  // 65536 latent vectors
constexpr float BETA_C   = 0.25f;

// output layout (flat concat, float32):
//   z_q_st [NVEC*DIM] | loss [1] | perplexity [1] | encodings [NVEC*TOPK*SIZE] | topk_idx [NVEC*TOPK]
constexpr int ZQ_ELEMS   = NVEC * DIM;                 // 4194304
constexpr int LOSS_OFF   = ZQ_ELEMS;                   // 4194304
constexpr int PERP_OFF   = ZQ_ELEMS + 1;               // 4194305
constexpr int ENC_OFF    = ZQ_ELEMS + 2;               // 4194306
constexpr long ENC_ELEMS = (long)NVEC * TOPK * SIZE;   // 201326592
constexpr int IDX_OFF    = ENC_OFF + (int)ENC_ELEMS;   // 205520898

typedef __attribute__((ext_vector_type(2))) float v2f;
typedef __attribute__((ext_vector_type(4))) float v4f;
typedef __attribute__((ext_vector_type(8))) float v8f;

// ---- branchless sorted-top3 insert (strict less: earliest index wins ties
//      automatically when candidates arrive in increasing-index order) ----
__device__ __forceinline__ void ins_fast(float& v0, float& v1, float& v2,
                                         int& i0, int& i1, int& i2,
                                         float cv, int ci) {
    const bool c0 = cv < v0;
    const bool c1 = cv < v1;
    const bool c2 = cv < v2;
    v2 = c1 ? v1 : (c2 ? cv : v2);
    i2 = c1 ? i1 : (c2 ? ci : i2);
    v1 = c0 ? v0 : (c1 ? cv : v1);
    i1 = c0 ? i0 : (c1 ? ci : i1);
    v0 = c0 ? cv : v0;
    i0 = c0 ? ci : i0;
}

// ---- branchless insert with lax.top_k tie-break (cross-lane merge only) ----
__device__ __forceinline__ void ins_tb(float& v0, float& v1, float& v2,
                                       int& i0, int& i1, int& i2,
                                       float cv, int ci) {
    const bool c0 = (cv < v0) || (cv == v0 && ci < i0);
    const bool c1 = (cv < v1) || (cv == v1 && ci < i1);
    const bool c2 = (cv < v2) || (cv == v2 && ci < i2);
    v2 = c1 ? v1 : (c2 ? cv : v2);
    i2 = c1 ? i1 : (c2 ? ci : i2);
    v1 = c0 ? v0 : (c1 ? cv : v1);
    i1 = c0 ? i0 : (c1 ? ci : i1);
    v0 = c0 ? cv : v0;
    i0 = c0 ? ci : i0;
}

// ---------------- prep: ||e||^2 + accumulator zeroing ----------------
__global__ void vq_prep(const float* __restrict__ emb, float* __restrict__ enorm,
                        float* __restrict__ lossAcc, unsigned* __restrict__ counts) {
    int t = blockIdx.x * blockDim.x + threadIdx.x;
    if (t < SIZE) {
        const float* e = emb + t * DIM;
        float s = 0.f;
        #pragma unroll
        for (int c = 0; c < DIM; ++c) s += e[c] * e[c];
        enorm[t]  = s;
        counts[t] = 0u;
        if (t == 0) *lossAcc = 0.f;
    }
}

// ---------------- zero the one-hot region ----------------
// base is 8B-aligned only (byte offset 16777224): head 2 floats, b128 body, tail 2.
__global__ void vq_zero(float* __restrict__ enc, long n) {
    const v4f z4 = {0.f, 0.f, 0.f, 0.f};
    v4f* p4 = (v4f*)(enc + 2);                 // 16B aligned
    const long n4 = (n - 4) >> 2;
    long i = (long)blockIdx.x * blockDim.x + threadIdx.x;
    const long stride = (long)gridDim.x * blockDim.x;
    for (; i < n4; i += stride)
        __builtin_nontemporal_store(z4, p4 + i);
    if (blockIdx.x == 0 && threadIdx.x == 0) {
        enc[0] = 0.f; enc[1] = 0.f;
        enc[n - 2] = 0.f; enc[n - 1] = 0.f;
    }
}

// ---- load 16 B-fragments (one 16x64 codebook tile, f32 16x16x4 layout) ----
__device__ __forceinline__ void loadB(v2f (&b)[16], const float* __restrict__ ebase) {
    #pragma unroll
    for (int kt = 0; kt < 16; ++kt)
        b[kt] = *(const v2f*)(ebase + kt * 4);
}

// ---- one column tile: prefetch next B, 16 chained WMMAs, top-3 scan ----
// A carries -2*z, accumulator is seeded with ||e_col||^2, so acc[r] == d - ||z_row||^2.
__device__ __forceinline__ void mm_step(const v2f (&a)[16], const v2f (&bc)[16],
                                        v2f (&bn)[16], bool pref,
                                        const float* __restrict__ ebnext,
                                        float en, int col,
                                        float (&tv)[8][3], int (&ti)[8][3]) {
    if (pref) loadB(bn, ebnext);               // overlaps with the WMMA chain below
    v8f acc = {en, en, en, en, en, en, en, en};
    #pragma unroll
    for (int kt = 0; kt < 16; ++kt)
        acc = __builtin_amdgcn_wmma_f32_16x16x4_f32(
            false, a[kt], false, bc[kt], (short)0, acc, false, false);
    #pragma unroll
    for (int r = 0; r < 8; ++r)
        ins_fast(tv[r][0], tv[r][1], tv[r][2], ti[r][0], ti[r][1], ti[r][2],
                 acc[r], col);
}

// ---------------- fused distance-GEMM + top-3 + epilogue ----------------
__global__ void __launch_bounds__(256)
vq_main(const float* __restrict__ z, const float* __restrict__ emb,
        float* __restrict__ out, const float* __restrict__ enorm,
        float* __restrict__ lossAcc, unsigned* __restrict__ counts) {
    const int lane = threadIdx.x & 31;
    const int wave = threadIdx.x >> 5;
    const int tile = blockIdx.x * 8 + wave;     // 0..4095
    const int rowBase = tile * 16;
    const int nl  = lane & 15;                  // M for A-frag, N for B/C-frag
    const int sub = lane >> 4;                  // half-wave selector

    float* out_zq  = out;
    float* out_enc = out + ENC_OFF;
    float* out_idx = out + IDX_OFF;

    // A fragments hold -2 * z-tile (16x64), f32 16x16x4 layout:
    // lane<16: M=lane, K = kt*4 + {0,1}; lane>=16: M=lane-16, K = kt*4 + {2,3}
    v2f a[16];
    const float* zrow = z + (rowBase + nl) * DIM + sub * 2;
    #pragma unroll
    for (int kt = 0; kt < 16; ++kt) {
        v2f t = *(const v2f*)(zrow + kt * 4);
        a[kt].x = -2.0f * t.x;
        a[kt].y = -2.0f * t.y;
    }

    // per-lane running top-3 per accumulator register (8 rows per half-wave)
    float tv[8][3]; int ti[8][3];
    #pragma unroll
    for (int r = 0; r < 8; ++r)
        #pragma unroll
        for (int j = 0; j < 3; ++j) { tv[r][j] = 3.402823466e38f; ti[r][j] = 0x7fffffff; }

    // --- sweep 64 column tiles, software-pipelined B double-buffer ---
    const float* ebase = emb + nl * DIM + sub * 2;
    v2f b0[16], b1[16];
    loadB(b0, ebase);
    for (int ct = 0; ct < 64; ct += 2) {
        const float en0 = enorm[ct * 16 + nl];
        const float en1 = enorm[(ct + 1) * 16 + nl];
        mm_step(a, b0, b1, true,          ebase + (ct + 1) * 16 * DIM, en0,
                ct * 16 + nl, tv, ti);
        mm_step(a, b1, b0, (ct + 2) < 64, ebase + (ct + 2) * 16 * DIM, en1,
                (ct + 1) * 16 + nl, tv, ti);
    }

    // --- cross-lane top-3 merge within each 16-lane half (rows are lane-striped) ---
    #pragma unroll
    for (int m = 1; m < 16; m <<= 1) {
        #pragma unroll
        for (int r = 0; r < 8; ++r) {
            float ov0 = __shfl_xor(tv[r][0], m, 32);
            float ov1 = __shfl_xor(tv[r][1], m, 32);
            float ov2 = __shfl_xor(tv[r][2], m, 32);
            int   oi0 = __shfl_xor(ti[r][0], m, 32);
            int   oi1 = __shfl_xor(ti[r][1], m, 32);
            int   oi2 = __shfl_xor(ti[r][2], m, 32);
            ins_tb(tv[r][0], tv[r][1], tv[r][2], ti[r][0], ti[r][1], ti[r][2], ov0, oi0);
            ins_tb(tv[r][0], tv[r][1], tv[r][2], ti[r][0], ti[r][1], ti[r][2], ov1, oi1);
            ins_tb(tv[r][0], tv[r][1], tv[r][2], ti[r][0], ti[r][1], ti[r][2], ov2, oi2);
        }
    }

    // --- epilogue: z_q (== z_q_st numerically), loss partial, one-hot, idx, counts ---
    float lsum = 0.f;
    #pragma unroll
    for (int r = 0; r < 8; ++r) {
        const int row = rowBase + sub * 8 + r;
        const int i0 = ti[r][0], i1 = ti[r][1], i2 = ti[r][2];
        const float* e0 = emb + i0 * DIM;
        const float* e1 = emb + i1 * DIM;
        const float* e2 = emb + i2 * DIM;
        #pragma unroll
        for (int cc = 0; cc < 4; ++cc) {
            const int c = nl + cc * 16;                 // coalesced across lanes
            float zq = (e0[c] + e1[c] + e2[c]) * (1.0f / 3.0f);
            float zz = z[row * DIM + c];
            out_zq[row * DIM + c] = zq;
            float df = zq - zz;
            lsum = fmaf(df, df, lsum);
        }
        if (nl == 0) {  // one lane per half writes per-row scalars
            out_idx[row * 3 + 0] = (float)i0;
            out_idx[row * 3 + 1] = (float)i1;
            out_idx[row * 3 + 2] = (float)i2;
            out_enc[(long)(row * 3 + 0) * SIZE + i0] = 1.0f;
            out_enc[(long)(row * 3 + 1) * SIZE + i1] = 1.0f;
            out_enc[(long)(row * 3 + 2) * SIZE + i2] = 1.0f;
            atomicAdd(&counts[i0], 1u);
            atomicAdd(&counts[i1], 1u);
            atomicAdd(&counts[i2], 1u);
        }
    }
    // wave-wide loss reduction, one atomic per wave
    #pragma unroll
    for (int m = 16; m >= 1; m >>= 1) lsum += __shfl_xor(lsum, m, 32);
    if (lane == 0) atomicAdd(lossAcc, lsum);
}

// ---------------- finalize: loss scalar + perplexity ----------------
__global__ void vq_fin(const unsigned* __restrict__ counts,
                       const float* __restrict__ lossAcc, float* __restrict__ out) {
    __shared__ float red[256];
    const int t = threadIdx.x;
    float s = 0.f;
    for (int i = t; i < SIZE; i += 256) {
        float em = (float)counts[i] * (1.0f / (float)(NVEC * TOPK));
        s += em * logf(em + 1e-10f);
    }
    red[t] = s; __syncthreads();
    for (int off = 128; off > 0; off >>= 1) {
        if (t < off) red[t] += red[t + off];
        __syncthreads();
    }
    if (t == 0) {
        // non-legacy loss: beta*mean(..) + mean(..) == (1+beta)*MSE
        out[LOSS_OFF] = (*lossAcc) * (1.0f + BETA_C) / (float)ZQ_ELEMS;
        out[PERP_OFF] = expf(-red[0]);
    }
}

// ---------------- entry ----------------
extern "C" void kernel_launch(void* const* d_in, const int* in_sizes, int n_in,
                              void* d_out, int out_size, void* d_ws, size_t ws_size,
                              hipStream_t stream) {
    const float* z   = (const float*)d_in[0];   // [65536,64]
    const float* emb = (const float*)d_in[1];   // [1024,64]
    float* out = (float*)d_out;

    // workspace: lossAcc(1 @0) | counts(1024 u32 @+8) | enorm(1024 f32 @+8+1024)
    float*    ws      = (float*)d_ws;
    float*    lossAcc = ws;
    unsigned* counts  = (unsigned*)(ws + 8);
    float*    enorm   = ws + 8 + SIZE;

    vq_prep<<<4, 256, 0, stream>>>(emb, enorm, lossAcc, counts);
    vq_zero<<<8192, 256, 0, stream>>>(out + ENC_OFF, ENC_ELEMS);
    vq_main<<<NVEC / 16 / 8, 256, 0, stream>>>(z, emb, out, enorm, lossAcc, counts);
    vq_fin<<<1, 256, 0, stream>>>(counts, lossAcc, out);
}